// MoeBlock_68650757260119
// MI455X (gfx1250) — compile-verified
//
#include <hip/hip_runtime.h>
#include <math.h>

typedef float v2f __attribute__((ext_vector_type(2)));
typedef float v8f __attribute__((ext_vector_type(8)));

#define NUM_EXPERTS 8
#define DIM 1024
#define HIDDEN 4096
#define TOKENS 8192           // B*S = 4*2048
#define MTILE 32              // tokens per block (2 WMMA token tiles)
#define XS_STRIDE 1028        // 1024 + 4 pad -> rows land on different LDS banks
#define HS_STRIDE 132         // 128 + 4 pad

// ---------------------------------------------------------------------------
// Router: logits = x @ wg^T + bg ; top-2 ; softmax over the two kept logits.
// One wave (32 lanes) per token. Writes full weights[T][8] (zeros off top-2).
// ---------------------------------------------------------------------------
__global__ __launch_bounds__(256) void moe_router_kernel(
    const float* __restrict__ x, const float* __restrict__ wg,
    const float* __restrict__ bg, float* __restrict__ weights) {
  const int wave = threadIdx.x >> 5;
  const int lane = threadIdx.x & 31;
  const int t = blockIdx.x * 8 + wave;
  const float* xr = x + (size_t)t * DIM;

  float acc[NUM_EXPERTS];
#pragma unroll
  for (int e = 0; e < NUM_EXPERTS; ++e) acc[e] = 0.f;

  for (int k = lane; k < DIM; k += 32) {
    const float xv = xr[k];
#pragma unroll
    for (int e = 0; e < NUM_EXPERTS; ++e) acc[e] += xv * wg[e * DIM + k];
  }
  // wave32 butterfly reduction -> every lane holds all 8 full sums
#pragma unroll
  for (int off = 16; off > 0; off >>= 1) {
#pragma unroll
    for (int e = 0; e < NUM_EXPERTS; ++e) acc[e] += __shfl_xor(acc[e], off, 32);
  }
#pragma unroll
  for (int e = 0; e < NUM_EXPERTS; ++e) acc[e] += bg[e];

  // top-2, lowest index wins ties (matches lax.top_k)
  int i1 = 0; float v1 = acc[0];
#pragma unroll
  for (int e = 1; e < NUM_EXPERTS; ++e) { if (acc[e] > v1) { v1 = acc[e]; i1 = e; } }
  int i2 = -1; float v2 = -3.0e38f;
#pragma unroll
  for (int e = 0; e < NUM_EXPERTS; ++e) { if (e != i1 && acc[e] > v2) { v2 = acc[e]; i2 = e; } }

  const float e1 = 1.0f;                 // exp(v1 - v1)
  const float e2 = __expf(v2 - v1);
  const float inv = 1.0f / (e1 + e2);
  if (lane < NUM_EXPERTS) {
    float wv = (lane == i1) ? e1 * inv : (lane == i2) ? e2 * inv : 0.f;
    weights[(size_t)t * NUM_EXPERTS + lane] = wv;
  }
}

// ---------------------------------------------------------------------------
// Fused expert FFN. Block = 32 tokens (2 WMMA token tiles), 8 waves. Loops all
// 8 experts (weights are 0 off the top-2, matching the reference exactly).
// Per hidden-block of 128: GEMM1 (g,u) for both token tiles sharing every
// weight fragment, SiLU*up*router-weight into LDS, GEMM2 accumulates into
// persistent out tiles (wave owns 128 out columns x 32 tokens). Deterministic.
// ---------------------------------------------------------------------------
__global__ __launch_bounds__(256) void moe_expert_kernel(
    const float* __restrict__ x, const float* __restrict__ Wgate,
    const float* __restrict__ Wup, const float* __restrict__ Wdown,
    const float* __restrict__ weights, float* __restrict__ out) {
  __shared__ float xs[MTILE * XS_STRIDE];        // ~128.5 KB  x tile
  __shared__ float hs[MTILE * HS_STRIDE];        // ~16.5 KB   h chunk
  __shared__ float wts[MTILE * NUM_EXPERTS];     // 1 KB router weights

  const int tid  = threadIdx.x;
  const int wave = tid >> 5;
  const int lane = tid & 31;
  const int m16  = lane & 15;   // M (or N-row) selector per WMMA layout
  const int hiL  = lane >> 4;   // half-wave -> K-pair / +8 row offset
  const int t0   = blockIdx.x * MTILE;

  // stage x tile into LDS (padded rows), float4 per thread
  for (int idx = tid; idx < MTILE * (DIM / 4); idx += 256) {
    const int r = idx >> 8;          // row 0..31
    const int c4 = idx & 255;        // float4 col
    ((float4*)(xs + r * XS_STRIDE))[c4] =
        ((const float4*)(x + (size_t)(t0 + r) * DIM))[c4];
  }
  if (tid < MTILE * NUM_EXPERTS)
    wts[tid] = weights[(size_t)t0 * NUM_EXPERTS + tid];
  __syncthreads();

  const v8f zero8 = {0.f, 0.f, 0.f, 0.f, 0.f, 0.f, 0.f, 0.f};
  v8f acc0[8], acc1[8];
#pragma unroll
  for (int i = 0; i < 8; ++i) { acc0[i] = zero8; acc1[i] = zero8; }

  for (int e = 0; e < NUM_EXPERTS; ++e) {
    const float* Wg = Wgate + (size_t)e * HIDDEN * DIM;
    const float* Wu = Wup   + (size_t)e * HIDDEN * DIM;
    const float* Wd = Wdown + (size_t)e * DIM * HIDDEN;

    for (int hb = 0; hb < HIDDEN; hb += 128) {
      // -------- GEMM1: g,u[2 x 16 tok x 16 hid] over K = DIM; B shared ----
      v8f g0 = zero8, u0 = zero8, g1 = zero8, u1 = zero8;
      const int hrow = hb + wave * 16 + m16;             // this lane's B row
      const float* wgrow = Wg + (size_t)hrow * DIM;
      const float* wurow = Wu + (size_t)hrow * DIM;
      const float* xrow0 = xs + m16 * XS_STRIDE;
      const float* xrow1 = xs + (16 + m16) * XS_STRIDE;
      for (int k0 = 0; k0 < DIM; k0 += 4) {
        const int c = k0 + 2 * hiL;                      // K-pair per layout
        v2f a0  = *(const v2f*)&xrow0[c];                // ds_load_b64
        v2f a1  = *(const v2f*)&xrow1[c];
        v2f bgv = *(const v2f*)&wgrow[c];                // global b64 (L2 hit)
        v2f buv = *(const v2f*)&wurow[c];
        g0 = __builtin_amdgcn_wmma_f32_16x16x4_f32(false, a0, false, bgv,
                                                   (short)0, g0, false, false);
        u0 = __builtin_amdgcn_wmma_f32_16x16x4_f32(false, a0, false, buv,
                                                   (short)0, u0, false, false);
        g1 = __builtin_amdgcn_wmma_f32_16x16x4_f32(false, a1, false, bgv,
                                                   (short)0, g1, false, false);
        u1 = __builtin_amdgcn_wmma_f32_16x16x4_f32(false, a1, false, buv,
                                                   (short)0, u1, false, false);
      }
      // SiLU(g)*u scaled by router weight (row scale commutes through the
      // down projection). C/D layout: token M = i + 8*hiL, hidden N = m16.
#pragma unroll
      for (int i = 0; i < 8; ++i) {
        const int m = i + 8 * hiL;
        const float ga = g0[i];
        const float sa = __builtin_amdgcn_rcpf(1.f + __expf(-ga));
        hs[m * HS_STRIDE + wave * 16 + m16] =
            ga * sa * u0[i] * wts[m * NUM_EXPERTS + e];
        const float gb = g1[i];
        const float sb = __builtin_amdgcn_rcpf(1.f + __expf(-gb));
        hs[(16 + m) * HS_STRIDE + wave * 16 + m16] =
            gb * sb * u1[i] * wts[(16 + m) * NUM_EXPERTS + e];
      }
      __syncthreads();
      // -------- GEMM2: out[:, wave*128..+127] += h @ Wd_chunk^T -----------
      const float* hrow0 = hs + m16 * HS_STRIDE;
      const float* hrow1 = hs + (16 + m16) * HS_STRIDE;
      for (int k0 = 0; k0 < 128; k0 += 4) {
        const int c = k0 + 2 * hiL;
        v2f a0 = *(const v2f*)&hrow0[c];                 // ds_load_b64
        v2f a1 = *(const v2f*)&hrow1[c];
#pragma unroll
        for (int nt = 0; nt < 8; ++nt) {
          const int nrow = wave * 128 + nt * 16 + m16;   // Wdown row (= out col)
          v2f b = *(const v2f*)&Wd[(size_t)nrow * HIDDEN + hb + c];
          acc0[nt] = __builtin_amdgcn_wmma_f32_16x16x4_f32(
              false, a0, false, b, (short)0, acc0[nt], false, false);
          acc1[nt] = __builtin_amdgcn_wmma_f32_16x16x4_f32(
              false, a1, false, b, (short)0, acc1[nt], false, false);
        }
      }
      __syncthreads();   // protect hs before next hidden-block rewrites it
    }
  }

  // store: (tile nt, vgpr i, lane) -> token i+8*hiL (+16), col wave*128+nt*16+m16
#pragma unroll
  for (int nt = 0; nt < 8; ++nt) {
#pragma unroll
    for (int i = 0; i < 8; ++i) {
      const int m = i + 8 * hiL;
      const int col = wave * 128 + nt * 16 + m16;
      out[(size_t)(t0 + m) * DIM + col] = acc0[nt][i];
      out[(size_t)(t0 + 16 + m) * DIM + col] = acc1[nt][i];
    }
  }
}

// ---------------------------------------------------------------------------
extern "C" void kernel_launch(void* const* d_in, const int* in_sizes, int n_in,
                              void* d_out, int out_size, void* d_ws, size_t ws_size,
                              hipStream_t stream) {
  const float* x     = (const float*)d_in[0];  // [4,2048,1024]
  const float* wg    = (const float*)d_in[1];  // [8,1024]
  const float* bg    = (const float*)d_in[2];  // [8]
  const float* Wgate = (const float*)d_in[3];  // [8,4096,1024]
  const float* Wup   = (const float*)d_in[4];  // [8,4096,1024]
  const float* Wdown = (const float*)d_in[5];  // [8,1024,4096]
  float* out = (float*)d_out;                  // [4,2048,1024]
  float* weights = (float*)d_ws;               // [8192,8] = 256 KB scratch

  moe_router_kernel<<<TOKENS / 8, 256, 0, stream>>>(x, wg, bg, weights);
  moe_expert_kernel<<<TOKENS / MTILE, 256, 0, stream>>>(x, Wgate, Wup, Wdown,
                                                        weights, out);
}